// Net_23931557773462
// MI455X (gfx1250) — compile-verified
//
#include <hip/hip_runtime.h>

#define N_NODES   100000
#define N_EDGES   600000
#define IN_FEATS  16
#define H_FEAT    128
#define E_FEATS   20
#define NUM_CLASSES 4
#define NUM_LAYERS  20
#define NEG_SLOPE 0.01f

typedef __attribute__((ext_vector_type(2))) float v2f;
typedef __attribute__((ext_vector_type(8))) float v8f;

// Monotonic float<->uint bijection: max over encoded uints == max over floats.
// Encoded 0 can only come from -NaN, so it is a safe "-inf / empty" sentinel.
__device__ __forceinline__ unsigned flipf(float f) {
    unsigned u = __float_as_uint(f);
    return (u & 0x80000000u) ? ~u : (u | 0x80000000u);
}
__device__ __forceinline__ float unflipf(unsigned u) {
    return __uint_as_float((u & 0x80000000u) ? (u ^ 0x80000000u) : ~u);
}

__global__ void k_init_agg(unsigned* __restrict__ agg, int n) {
    int i = blockIdx.x * blockDim.x + threadIdx.x;
    if (i < n) agg[i] = 0u;
}

// One thread per (edge, 4-feature chunk): gather h[src]*w, native u32 atomic max into agg[dst].
template <int F>
__global__ void k_edge_max(const float* __restrict__ h,
                           const float* __restrict__ edge_feat,
                           const int* __restrict__ src,
                           const int* __restrict__ dst,
                           unsigned* __restrict__ agg) {
    const int CH = F / 4;
    int tid = blockIdx.x * blockDim.x + threadIdx.x;
    if (tid >= N_EDGES * CH) return;
    int e = tid / CH;
    int p = tid % CH;
    int s = src[e];
    int d = dst[e];
    float w = edge_feat[e * E_FEATS];          // w_e = edge_feat[:,0]
    const float4 hv = *(const float4*)(h + s * F + p * 4);
    unsigned* ap = agg + d * F + p * 4;
    atomicMax(ap + 0, flipf(hv.x * w));
    atomicMax(ap + 1, flipf(hv.y * w));
    atomicMax(ap + 2, flipf(hv.z * w));
    atomicMax(ap + 3, flipf(hv.w * w));
}

// Block = 256 threads = 8 waves. Block handles 16 nodes; wave w handles output
// columns [16w, 16w+16). fp32 WMMA 16x16x4, K-loop over in-feats.
template <int K, bool RELU>
__global__ void k_gemm_wmma(const float* __restrict__ hin,     // N x K
                            const unsigned* __restrict__ agg,  // N x K (flipped)
                            const float* __restrict__ W,       // K x H_FEAT row-major
                            const float* __restrict__ b,       // H_FEAT
                            float* __restrict__ hout) {        // N x H_FEAT
    __shared__ float rst[16][K + 4];   // +4 pad -> conflict-free column reads
    const int base = blockIdx.x * 16;

    // Stage rst = h + decode(agg) for this 16-node tile.
    for (int idx = threadIdx.x; idx < 16 * K; idx += blockDim.x) {
        int r = idx / K, k = idx % K;
        int node = base + r;
        unsigned ab = agg[node * K + k];
        float a = (ab == 0u) ? 0.0f : unflipf(ab);   // empty segment -> 0
        rst[r][k] = hin[node * K + k] + a;
    }
    __syncthreads();

    const int wave  = threadIdx.x >> 5;        // 0..7 -> column tile
    const int lane  = threadIdx.x & 31;
    const int col   = (wave << 4) + (lane & 15);
    const int khalf = (lane >> 4) << 1;        // lanes 0-15: K=k0,k0+1 ; 16-31: K=k0+2,k0+3

    float bias = b[col];                       // D[m][n] bias depends on n only
    v8f c = { bias, bias, bias, bias, bias, bias, bias, bias };

#pragma unroll
    for (int k0 = 0; k0 < K; k0 += 4) {
        v2f a, bb;
        a.x  = rst[lane & 15][k0 + khalf];
        a.y  = rst[lane & 15][k0 + khalf + 1];
        bb.x = W[(k0 + khalf) * H_FEAT + col];
        bb.y = W[(k0 + khalf + 1) * H_FEAT + col];
        c = __builtin_amdgcn_wmma_f32_16x16x4_f32(false, a, false, bb,
                                                  (short)0, c, false, false);
    }

    const int moff = (lane >> 4) << 3;         // lanes 0-15: M=i ; lanes 16-31: M=i+8
#pragma unroll
    for (int i = 0; i < 8; i++) {
        float v = c[i];
        if (RELU) v = (v > 0.0f) ? v : NEG_SLOPE * v;
        hout[(base + moff + i) * H_FEAT + col] = v;
    }
}

// Final 128 -> 4 projection (tiny, VALU).
__global__ void k_final(const float* __restrict__ h,
                        const float* __restrict__ Wf,   // 128 x 4 row-major
                        const float* __restrict__ bf,
                        float* __restrict__ out) {
    int n = blockIdx.x * blockDim.x + threadIdx.x;
    if (n >= N_NODES) return;
    float a0 = bf[0], a1 = bf[1], a2 = bf[2], a3 = bf[3];
    const float* hp = h + n * H_FEAT;
#pragma unroll 4
    for (int k = 0; k < H_FEAT; k++) {
        float hv = hp[k];
        const float4 w = *(const float4*)(Wf + k * 4);
        a0 += hv * w.x; a1 += hv * w.y; a2 += hv * w.z; a3 += hv * w.w;
    }
    float4 o = { a0, a1, a2, a3 };
    *(float4*)(out + n * 4) = o;
}

extern "C" void kernel_launch(void* const* d_in, const int* in_sizes, int n_in,
                              void* d_out, int out_size, void* d_ws, size_t ws_size,
                              hipStream_t stream) {
    const float* node_feat = (const float*)d_in[0];
    const float* edge_feat = (const float*)d_in[1];
    const int*   src       = (const int*)d_in[2];
    const int*   dst       = (const int*)d_in[3];
    const float* W1        = (const float*)d_in[4];
    const float* b1        = (const float*)d_in[5];
    const float* Ws        = (const float*)d_in[6];
    const float* bs        = (const float*)d_in[7];
    const float* Wf        = (const float*)d_in[8];
    const float* bf        = (const float*)d_in[9];

    char* ws = (char*)d_ws;
    const size_t hbytes = (size_t)N_NODES * H_FEAT * sizeof(float);  // 51.2 MB
    unsigned* agg = (unsigned*)ws;
    float* hA = (float*)(ws + hbytes);
    float* hB = (float*)(ws + 2 * hbytes);

    // ---- Layer 0 (16 -> 128) ----
    {
        int n = N_NODES * IN_FEATS;
        k_init_agg<<<(n + 255) / 256, 256, 0, stream>>>(agg, n);
        int t = N_EDGES * (IN_FEATS / 4);
        k_edge_max<IN_FEATS><<<(t + 255) / 256, 256, 0, stream>>>(
            node_feat, edge_feat, src, dst, agg);
        k_gemm_wmma<IN_FEATS, true><<<N_NODES / 16, 256, 0, stream>>>(
            node_feat, agg, W1, b1, hA);
    }

    // ---- Layers 1..19 (128 -> 128) ----
    float* cur = hA;
    float* nxt = hB;
    for (int l = 0; l < NUM_LAYERS - 1; l++) {
        const float* W  = Ws + (size_t)l * H_FEAT * H_FEAT;
        const float* bb = bs + (size_t)l * H_FEAT;
        int n = N_NODES * H_FEAT;
        k_init_agg<<<(n + 255) / 256, 256, 0, stream>>>(agg, n);
        int t = N_EDGES * (H_FEAT / 4);
        k_edge_max<H_FEAT><<<(t + 255) / 256, 256, 0, stream>>>(
            cur, edge_feat, src, dst, agg);
        k_gemm_wmma<H_FEAT, true><<<N_NODES / 16, 256, 0, stream>>>(
            cur, agg, W, bb, nxt);
        float* tmp = cur; cur = nxt; nxt = tmp;
    }

    // ---- Final projection ----
    k_final<<<(N_NODES + 255) / 256, 256, 0, stream>>>(cur, Wf, bf, (float*)d_out);
}